// hipRSSM_48412871360734
// MI455X (gfx1250) — compile-verified
//
#include <hip/hip_runtime.h>
#include <hip/hip_bf16.h>
#include <math.h>

// Problem constants (from reference): B=512, T=128
#define B_   512
#define T_   128
#define OBS_ 32
#define ACT_ 8
#define HID_ 128
#define LOD_ 64
#define LSD_ 128
#define K_   15
#define BT_  (B_*T_)

typedef __attribute__((ext_vector_type(2))) float v2f;
typedef __attribute__((ext_vector_type(8))) float v8f;

__device__ __forceinline__ float elup1f(float x) {
  // elu(x)+1 : x>0 -> x+1 ; else exp(x)
  return x > 0.f ? x + 1.f : __expf(x);
}

// ---- WMMA f32 16x16x4 helpers (CDNA5) --------------------------------------
__device__ __forceinline__ v8f wmma4(v2f a, v2f b, v8f c) {
  // (neg_a, A, neg_b, B, c_mod, C, reuse_a, reuse_b)
  return __builtin_amdgcn_wmma_f32_16x16x4_f32(false, a, false, b, (short)0, c,
                                               false, false);
}

// A fragment: row-major M(16) x K, lane m = lane&15, k pair = kb + 2*(lane>=16)
__device__ __forceinline__ v2f ldA(const float* A, int lda, int kb) {
  int l = threadIdx.x & 31;
  int m = l & 15;
  int k = kb + ((l >> 4) << 1);
  v2f a; a.x = A[m * lda + k]; a.y = A[m * lda + k + 1];
  return a;
}
// B fragment: row-major K x N, lane n = nb + (lane&15), k pair as above
__device__ __forceinline__ v2f ldB(const float* Bm, int ldb, int kb, int nb) {
  int l = threadIdx.x & 31;
  int n = nb + (l & 15);
  int k = kb + ((l >> 4) << 1);
  v2f b; b.x = Bm[k * ldb + n]; b.y = Bm[(k + 1) * ldb + n];
  return b;
}
// B fragment from k-pair-packed weights: Wp[((k>>1)*ldn + n)*2 + (k&1)]
// -> single contiguous 8B load per fragment
__device__ __forceinline__ v2f ldBp(const float* Wp, int ldn, int kb, int nb) {
  int l = threadIdx.x & 31;
  int k = kb + ((l >> 4) << 1);
  const float* p = Wp + ((size_t)(k >> 1) * ldn + nb + (l & 15)) * 2;
  v2f b; b.x = p[0]; b.y = p[1];
  return b;
}

__device__ __forceinline__ v8f gemm_tile(const float* A, int lda,
                                         const float* Bm, int ldb,
                                         int K, int nb, v8f acc) {
  #pragma unroll 4
  for (int k = 0; k < K; k += 4)
    acc = wmma4(ldA(A, lda, k), ldB(Bm, ldb, k, nb), acc);
  return acc;
}

// D layout: element r of v8f -> row m = r + 8*(lane>=16), col n = nb + (lane&15)

// ---- workspace init ---------------------------------------------------------
__global__ void k_init(float* prec, float* accm) {
  int i = blockIdx.x * blockDim.x + threadIdx.x;
  if (i < B_ * LSD_) { prec[i] = 1.f; accm[i] = 0.f; }
}

// Repack tm11..tm22 (15,64,64) into WMMA-B fragment order with K padded to 16.
// tmB[(m*8 + s*2 + h)*8192 + f*2 + c] = tm_m[k=4s+2h+c][f], zero for k==15.
__global__ void k_prep(const float* tm11, const float* tm12,
                       const float* tm21, const float* tm22, float* tmB) {
  int idx = blockIdx.x * blockDim.x + threadIdx.x;
  if (idx >= 4 * 16 * 4096) return;
  int f = idx & 4095;
  int k = (idx >> 12) & 15;
  int m = idx >> 16;
  const float* src = (m == 0) ? tm11 : (m == 1) ? tm12 : (m == 2) ? tm21 : tm22;
  float v = (k < K_) ? src[k * 4096 + f] : 0.f;
  int s = k >> 2, h = (k >> 1) & 1, c = k & 1;
  tmB[(size_t)(m * 8 + s * 2 + h) * 8192 + (size_t)f * 2 + c] = v;
}

// Generic K x N weight repack into k-pair layout (N padded to Npad, zeros).
__global__ void k_packw(const float* src, float* dst, int K, int N, int Npad) {
  int idx = blockIdx.x * blockDim.x + threadIdx.x;
  if (idx >= K * Npad) return;
  int k = idx / Npad, n = idx % Npad;
  float v = (n < N) ? src[k * N + n] : 0.f;
  dst[((size_t)(k >> 1) * Npad + n) * 2 + (k & 1)] = v;
}

// ---- encoders: he->w_mean/w_var and act->ctrl, 16 rows per block -----------
__global__ __launch_bounds__(128) void k_enc(
    const float* obs, const float* act,
    const float* We1, const float* be1, const float* Wem, const float* bem,
    const float* Wev, const float* bev, const float* Wu1, const float* bu1,
    const float* Wu2, const float* bu2,
    float* wmean, float* wvar, float* ctrl) {
  __shared__ float sIn[16 * OBS_];
  __shared__ float sAct[16 * ACT_];
  __shared__ float sH[16 * HID_];
  const int tid = threadIdx.x, l = tid & 31, w = tid >> 5;
  const int row0 = blockIdx.x * 16;

  for (int e = tid; e < 16 * OBS_; e += 128) sIn[e] = obs[row0 * OBS_ + e];
  if (tid < 16 * ACT_) sAct[tid] = act[row0 * ACT_ + tid];
  __syncthreads();

  // he = relu(obs @ We1 + be1)   (16x128)
  for (int nt = w; nt < 8; nt += 4) {
    v8f d = {};
    d = gemm_tile(sIn, OBS_, We1, HID_, OBS_, nt * 16, d);
    int n = nt * 16 + (l & 15);
    #pragma unroll
    for (int r = 0; r < 8; ++r) {
      int m = r + ((l >> 4) << 3);
      float v = d[r] + be1[n];
      sH[m * HID_ + n] = v > 0.f ? v : 0.f;
    }
  }
  __syncthreads();

  // w_mean = he@Wem+bem, w_var = elup1(he@Wev+bev)   (16x64 each)
  {
    v8f dm = {}, dv = {};
    dm = gemm_tile(sH, HID_, Wem, LOD_, HID_, w * 16, dm);
    dv = gemm_tile(sH, HID_, Wev, LOD_, HID_, w * 16, dv);
    int n = w * 16 + (l & 15);
    #pragma unroll
    for (int r = 0; r < 8; ++r) {
      int m = r + ((l >> 4) << 3);
      wmean[(size_t)(row0 + m) * LOD_ + n] = dm[r] + bem[n];
      wvar [(size_t)(row0 + m) * LOD_ + n] = elup1f(dv[r] + bev[n]);
    }
  }
  __syncthreads();

  // h2 = relu(act @ Wu1 + bu1)
  for (int nt = w; nt < 8; nt += 4) {
    v8f d = {};
    d = gemm_tile(sAct, ACT_, Wu1, HID_, ACT_, nt * 16, d);
    int n = nt * 16 + (l & 15);
    #pragma unroll
    for (int r = 0; r < 8; ++r) {
      int m = r + ((l >> 4) << 3);
      float v = d[r] + bu1[n];
      sH[m * HID_ + n] = v > 0.f ? v : 0.f;
    }
  }
  __syncthreads();

  // ctrl = h2 @ Wu2 + bu2   (16x128)
  for (int nt = w; nt < 8; nt += 4) {
    v8f d = {};
    d = gemm_tile(sH, HID_, Wu2, HID_, HID_, nt * 16, d);
    int n = nt * 16 + (l & 15);
    #pragma unroll
    for (int r = 0; r < 8; ++r) {
      int m = r + ((l >> 4) << 3);
      ctrl[(size_t)(row0 + m) * LSD_ + n] = d[r] + bu2[n];
    }
  }
}

// ---- context: accumulate task precision / weighted mean --------------------
__global__ __launch_bounds__(128) void k_ctx(
    const float* obs, const float* act, const unsigned char* valid,
    const float* Wt1, const float* bt1, const float* Wtm, const float* btm,
    const float* Wtv, const float* btv, float* prec, float* accm) {
  __shared__ float sIn[16 * 72];
  __shared__ float sH[16 * HID_];
  const int tid = threadIdx.x, l = tid & 31, w = tid >> 5;
  const int rr0 = blockIdx.x * 16;   // rows over (b, t) with t in [0,63)

  for (int e = tid; e < 16 * 72; e += 128) {
    int m = e / 72, c = e % 72;
    int rr = rr0 + m, b = rr / 63, t = rr % 63;
    float v;
    if (c < 32)      v = obs[(size_t)(b * T_ + t) * OBS_ + c];
    else if (c < 40) v = act[(size_t)(b * T_ + t) * ACT_ + (c - 32)];
    else             v = obs[(size_t)(b * T_ + t + 1) * OBS_ + (c - 40)];
    sIn[e] = v;
  }
  __syncthreads();

  // h = relu(ctx @ Wt1 + bt1), K = 72
  for (int nt = w; nt < 8; nt += 4) {
    v8f d = {};
    d = gemm_tile(sIn, 72, Wt1, HID_, 72, nt * 16, d);
    int n = nt * 16 + (l & 15);
    #pragma unroll
    for (int r = 0; r < 8; ++r) {
      int m = r + ((l >> 4) << 3);
      float v = d[r] + bt1[n];
      sH[m * HID_ + n] = v > 0.f ? v : 0.f;
    }
  }
  __syncthreads();

  // tmean/tvar tiles + streamed atomic accumulation
  for (int nt = w; nt < 8; nt += 4) {
    v8f dm = {}, dv = {};
    dm = gemm_tile(sH, HID_, Wtm, LSD_, HID_, nt * 16, dm);
    dv = gemm_tile(sH, HID_, Wtv, LSD_, HID_, nt * 16, dv);
    int n = nt * 16 + (l & 15);
    #pragma unroll
    for (int r = 0; r < 8; ++r) {
      int m = r + ((l >> 4) << 3);
      int rr = rr0 + m, b = rr / 63, t = rr % 63;
      if (valid[b * T_ + t]) {
        float tv = elup1f(dv[r] + btv[n]);
        float tm = dm[r] + btm[n];
        atomicAdd(&prec[b * LSD_ + n], 1.f / tv);
        atomicAdd(&accm[b * LSD_ + n], tm / tv);
      }
    }
  }
}

// ---- task posterior finalize + noise --------------------------------------
__global__ __launch_bounds__(128) void k_task(
    const float* prec, const float* accm, const float* Wtn,
    const float* log_noise, float* taskm, float* noise) {
  __shared__ float stv[LSD_];
  int b = blockIdx.x, j = threadIdx.x;
  float tv = 1.f / prec[b * LSD_ + j];
  float tm = tv * accm[b * LSD_ + j];
  taskm[b * LSD_ + j] = tm;
  stv[j] = tv;
  __syncthreads();
  float s = 0.f;
  #pragma unroll 4
  for (int i = 0; i < LSD_; ++i) {
    float wv = Wtn[i * LSD_ + j];
    s += stv[i] * wv * wv;
  }
  noise[b * LSD_ + j] = elup1f(log_noise[j]) + s;
}

// ---- the sequential scan: 16 batch elems per block, 16 waves ---------------
__global__ __launch_bounds__(512) void k_scan(
    const float* wmean, const float* wvar, const unsigned char* valid,
    const float* ctrl, const float* taskm_g, const float* noise_g,
    const float* Wc1p, const float* bc1, const float* Wc2p, const float* bc2,
    const float* tmB, float* priorm, float* priorc) {
  __shared__ float sMu[16][LSD_];
  __shared__ float sCu[16][LOD_], sCl[16][LOD_], sCs[16][LOD_];
  __shared__ float sNMu[16][LSD_];
  __shared__ float sNCu[16][LOD_], sNCl[16][LOD_], sNCs[16][LOD_];
  __shared__ float sTask[16][LSD_];
  __shared__ float sH[16][HID_];
  __shared__ float sCoef[16][16];

  // sNMu / sNCu are dead during phase (b): reuse as partial-sum buffers.
  float* sH2 = &sNMu[0][0];   // 16x128 second-half hidden partial
  float* sLg = &sNCu[0][0];   // 4 x (16x16) logits partials

  const int tid = threadIdx.x, l = tid & 31, w = tid >> 5;
  const int b0 = blockIdx.x * 16;

  for (int e = tid; e < 16 * LSD_; e += 512) {
    int b = e >> 7, x = e & 127;
    sMu[b][x] = 0.f;
    sTask[b][x] = taskm_g[(b0 + b) * LSD_ + x];
  }
  for (int e = tid; e < 16 * LOD_; e += 512) {
    int b = e >> 6, j = e & 63;
    sCu[b][j] = 10.f; sCl[b][j] = 10.f; sCs[b][j] = 1.f;   // INIT_COVAR, cs=1
  }
  __syncthreads();

  for (int t = 0; t < T_; ++t) {
    // (a) Kalman update, in place; prefetch next step's observations
    for (int e = tid; e < 16 * LOD_; e += 512) {
      int b = e >> 6, j = e & 63;
      size_t row = ((size_t)(b0 + b) * T_ + t) * LOD_ + j;
      float wm = wmean[row];
      float wv = wvar [row];
      if (t + 1 < T_) {
        __builtin_prefetch(&wmean[row + LOD_], 0, 3);
        __builtin_prefetch(&wvar [row + LOD_], 0, 3);
      }
      bool  vl = valid[(b0 + b) * T_ + t] != 0;
      float cu = sCu[b][j], cl = sCl[b][j], cs = sCs[b][j];
      float muU = sMu[b][j], muL = sMu[b][LOD_ + j];
      float denom = cu + wv;
      float qu = cu / denom, ql = cs / denom;
      float res = wm - muU;
      if (vl) {
        sMu[b][j]        = muU + qu * res;
        sMu[b][LOD_ + j] = muL + ql * res;
        sCu[b][j] = (1.f - qu) * cu;
        sCl[b][j] = cl - ql * cs;
        sCs[b][j] = (1.f - qu) * cs;
      }
    }
    __syncthreads();

    // (b) mixture coefficients: relu([post_mu, task_mean] @ Wc1 + bc1) @ Wc2
    // K=256 split across wave halves: waves 0-7 do post_mu x Wc1[:128],
    // waves 8-15 do task_mean x Wc1[128:]; combine partials through LDS.
    {
      int nt = w & 7;
      const float* Ap = (w < 8) ? &sMu[0][0] : &sTask[0][0];
      const float* Bp = Wc1p + ((w < 8) ? 0 : (size_t)64 * HID_ * 2);
      v8f d = {};
      #pragma unroll 4
      for (int k = 0; k < 128; k += 4)
        d = wmma4(ldA(Ap, LSD_, k), ldBp(Bp, HID_, k, nt * 16), d);
      float* dst = (w < 8) ? &sH[0][0] : sH2;
      int n = nt * 16 + (l & 15);
      #pragma unroll
      for (int r = 0; r < 8; ++r) {
        int m = r + ((l >> 4) << 3);
        dst[m * HID_ + n] = d[r];
      }
    }
    __syncthreads();
    for (int e = tid; e < 16 * HID_; e += 512) {
      int n = e & 127;
      float v = (&sH[0][0])[e] + sH2[e] + bc1[n];
      (&sH[0][0])[e] = v > 0.f ? v : 0.f;
    }
    __syncthreads();
    // logits (N=15 padded to 16): K=128 split across 4 waves
    if (w < 4) {
      v8f d = {};
      #pragma unroll
      for (int k = 0; k < 32; k += 4) {
        int kk = w * 32 + k;
        d = wmma4(ldA(&sH[0][0], HID_, kk), ldBp(Wc2p, 16, kk, 0), d);
      }
      int n = l & 15;
      #pragma unroll
      for (int r = 0; r < 8; ++r) {
        int m = r + ((l >> 4) << 3);
        sLg[w * 256 + m * 16 + n] = d[r];
      }
    }
    __syncthreads();
    if (tid < 256) {
      int m = tid >> 4, n = tid & 15;
      float v = sLg[tid] + sLg[256 + tid] + sLg[512 + tid] + sLg[768 + tid];
      sCoef[m][n] = (n < K_) ? v + bc2[n] : 0.f;
    }
    __syncthreads();
    if (tid < 16) {                     // softmax over K=15
      float mx = -1e30f;
      for (int k = 0; k < K_; ++k) mx = fmaxf(mx, sCoef[tid][k]);
      float s = 0.f;
      for (int k = 0; k < K_; ++k) {
        float ev = __expf(sCoef[tid][k] - mx);
        sCoef[tid][k] = ev; s += ev;
      }
      float inv = 1.f / s;
      for (int k = 0; k < K_; ++k) sCoef[tid][k] *= inv;
      sCoef[tid][15] = 0.f;             // K padded to 16
    }
    __syncthreads();

    // (c) build T tiles with WMMA (A = coef 16x16, B = tm basis), consume
    v2f aF[4];
    #pragma unroll
    for (int s = 0; s < 4; ++s) aF[s] = ldA(&sCoef[0][0], 16, s * 4);

    for (int ii = 0; ii < 4; ++ii) {
      int i = (w << 2) + ii;            // output row of T handled by this wave
      float aMU[8], aML[8], aCU[8], aCL[8], aCS[8];
      #pragma unroll
      for (int r = 0; r < 8; ++r) { aMU[r]=aML[r]=aCU[r]=aCL[r]=aCS[r]=0.f; }

      for (int j0 = 0; j0 < LOD_; j0 += 16) {
        int nf = i * LOD_ + j0 + (l & 15);
        int h  = l >> 4;
        v8f d11 = {}, d12 = {}, d21 = {}, d22 = {};
        #pragma unroll
        for (int s = 0; s < 4; ++s) {
          const float* p0 = tmB + (size_t)(s * 2 + h) * 8192 + (size_t)nf * 2;
          v2f b11; b11.x = p0[0];      b11.y = p0[1];
          v2f b12; b12.x = p0[65536];  b12.y = p0[65537];
          v2f b21; b21.x = p0[131072]; b21.y = p0[131073];
          v2f b22; b22.x = p0[196608]; b22.y = p0[196609];
          d11 = wmma4(aF[s], b11, d11);
          d12 = wmma4(aF[s], b12, d12);
          d21 = wmma4(aF[s], b21, d21);
          d22 = wmma4(aF[s], b22, d22);
        }
        int jj = j0 + (l & 15);
        float diag = (jj == i) ? 1.f : 0.f;
        #pragma unroll
        for (int r = 0; r < 8; ++r) {
          int b = r + (h << 3);
          float muU = sMu[b][jj], muL = sMu[b][LOD_ + jj];
          float cu = sCu[b][jj], cl = sCl[b][jj], cs = sCs[b][jj];
          float t11 = d11[r] + diag;
          float t12 = d12[r];
          float t21 = d21[r];
          float t22 = d22[r] + diag;
          aMU[r] += t11 * muU + t12 * muL;
          aML[r] += t21 * muU + t22 * muL;
          aCU[r] += t11 * t11 * cu + t12 * t12 * cl + 2.f * t11 * t12 * cs;
          aCL[r] += t21 * t21 * cu + t22 * t22 * cl + 2.f * t21 * t22 * cs;
          aCS[r] += t21 * t11 * cu + t22 * t12 * cl + (t11 * t22 + t12 * t21) * cs;
        }
      }
      // reduce over the 16 lanes holding the contraction index j
      #pragma unroll
      for (int r = 0; r < 8; ++r) {
        #pragma unroll
        for (int m = 1; m < 16; m <<= 1) {
          aMU[r] += __shfl_xor(aMU[r], m, 16);
          aML[r] += __shfl_xor(aML[r], m, 16);
          aCU[r] += __shfl_xor(aCU[r], m, 16);
          aCL[r] += __shfl_xor(aCL[r], m, 16);
          aCS[r] += __shfl_xor(aCS[r], m, 16);
        }
      }
      if ((l & 15) == 0) {
        #pragma unroll
        for (int r = 0; r < 8; ++r) {
          int b = r + ((l >> 4) << 3);
          int gb = b0 + b;
          sNMu[b][i]        = aMU[r] + ctrl[((size_t)gb * T_ + t) * LSD_ + i];
          sNMu[b][LOD_ + i] = aML[r] + ctrl[((size_t)gb * T_ + t) * LSD_ + LOD_ + i];
          sNCu[b][i] = aCU[r] + noise_g[gb * LSD_ + i];
          sNCl[b][i] = aCL[r] + noise_g[gb * LSD_ + LOD_ + i];
          sNCs[b][i] = aCS[r];
        }
      }
    }
    __syncthreads();

    // (d) commit new state + emit prior mean/cov for this step
    for (int e = tid; e < 16 * LSD_; e += 512) {
      int b = e >> 7, x = e & 127;
      float v = sNMu[b][x];
      sMu[b][x] = v;
      priorm[((size_t)(b0 + b) * T_ + t) * LSD_ + x] = v;
    }
    for (int e = tid; e < 16 * 192; e += 512) {
      int b = e / 192, x = e % 192;
      float v;
      if (x < 64)       { v = sNCu[b][x];       sCu[b][x]       = v; }
      else if (x < 128) { v = sNCl[b][x - 64];  sCl[b][x - 64]  = v; }
      else              { v = sNCs[b][x - 128]; sCs[b][x - 128] = v; }
      priorc[((size_t)(b0 + b) * T_ + t) * 192 + x] = v;
    }
    __syncthreads();
  }
}

// ---- decoders ---------------------------------------------------------------
__global__ __launch_bounds__(128) void k_dec(
    const float* priorm, const float* priorc,
    const float* Wdm1, const float* bdm1, const float* Wdm2, const float* bdm2,
    const float* Wdv1, const float* bdv1, const float* Wdv2, const float* bdv2,
    float* out) {
  __shared__ float sPm[16 * LSD_];
  __shared__ float sPc[16 * 192];
  __shared__ float sH[16 * HID_];
  const int tid = threadIdx.x, l = tid & 31, w = tid >> 5;
  const int row0 = blockIdx.x * 16;

  for (int e = tid; e < 16 * LSD_; e += 128) sPm[e] = priorm[(size_t)row0 * LSD_ + e];
  for (int e = tid; e < 16 * 192;  e += 128) sPc[e] = priorc[(size_t)row0 * 192 + e];
  __syncthreads();

  // mean head
  for (int nt = w; nt < 8; nt += 4) {
    v8f d = {};
    d = gemm_tile(sPm, LSD_, Wdm1, HID_, LSD_, nt * 16, d);
    int n = nt * 16 + (l & 15);
    #pragma unroll
    for (int r = 0; r < 8; ++r) {
      int m = r + ((l >> 4) << 3);
      float v = d[r] + bdm1[n];
      sH[m * HID_ + n] = v > 0.f ? v : 0.f;
    }
  }
  __syncthreads();
  if (w < 2) {
    v8f d = {};
    d = gemm_tile(sH, HID_, Wdm2, OBS_, HID_, w * 16, d);
    int n = w * 16 + (l & 15);
    #pragma unroll
    for (int r = 0; r < 8; ++r) {
      int m = r + ((l >> 4) << 3);
      out[(size_t)(row0 + m) * OBS_ + n] = d[r] + bdm2[n];
    }
  }
  __syncthreads();

  // var head (K = 192)
  for (int nt = w; nt < 8; nt += 4) {
    v8f d = {};
    d = gemm_tile(sPc, 192, Wdv1, HID_, 192, nt * 16, d);
    int n = nt * 16 + (l & 15);
    #pragma unroll
    for (int r = 0; r < 8; ++r) {
      int m = r + ((l >> 4) << 3);
      float v = d[r] + bdv1[n];
      sH[m * HID_ + n] = v > 0.f ? v : 0.f;
    }
  }
  __syncthreads();
  if (w < 2) {
    v8f d = {};
    d = gemm_tile(sH, HID_, Wdv2, OBS_, HID_, w * 16, d);
    int n = w * 16 + (l & 15);
    #pragma unroll
    for (int r = 0; r < 8; ++r) {
      int m = r + ((l >> 4) << 3);
      out[(size_t)BT_ * OBS_ + (size_t)(row0 + m) * OBS_ + n] =
          elup1f(d[r] + bdv2[n]);
    }
  }
}

// ---- host launch ------------------------------------------------------------
extern "C" void kernel_launch(void* const* d_in, const int* in_sizes, int n_in,
                              void* d_out, int out_size, void* d_ws, size_t ws_size,
                              hipStream_t stream) {
  (void)in_sizes; (void)n_in; (void)out_size; (void)ws_size;
  const float* obs   = (const float*)d_in[0];
  const float* act   = (const float*)d_in[1];
  const unsigned char* valid = (const unsigned char*)d_in[2];  // jnp bool_
  // d_in[3] = context_len (== 64, compiled in)
  const float* We1 = (const float*)d_in[4];
  const float* be1 = (const float*)d_in[5];
  const float* Wem = (const float*)d_in[6];
  const float* bem = (const float*)d_in[7];
  const float* Wev = (const float*)d_in[8];
  const float* bev = (const float*)d_in[9];
  const float* Wt1 = (const float*)d_in[10];
  const float* bt1 = (const float*)d_in[11];
  const float* Wtm = (const float*)d_in[12];
  const float* btm = (const float*)d_in[13];
  const float* Wtv = (const float*)d_in[14];
  const float* btv = (const float*)d_in[15];
  const float* Wc1 = (const float*)d_in[16];
  const float* bc1 = (const float*)d_in[17];
  const float* Wc2 = (const float*)d_in[18];
  const float* bc2 = (const float*)d_in[19];
  const float* Wu1 = (const float*)d_in[20];
  const float* bu1 = (const float*)d_in[21];
  const float* Wu2 = (const float*)d_in[22];
  const float* bu2 = (const float*)d_in[23];
  const float* tm11 = (const float*)d_in[24];
  const float* tm12 = (const float*)d_in[25];
  const float* tm21 = (const float*)d_in[26];
  const float* tm22 = (const float*)d_in[27];
  const float* log_noise = (const float*)d_in[28];
  const float* Wtn  = (const float*)d_in[29];
  const float* Wdm1 = (const float*)d_in[30];
  const float* bdm1 = (const float*)d_in[31];
  const float* Wdm2 = (const float*)d_in[32];
  const float* bdm2 = (const float*)d_in[33];
  const float* Wdv1 = (const float*)d_in[34];
  const float* bdv1 = (const float*)d_in[35];
  const float* Wdv2 = (const float*)d_in[36];
  const float* bdv2 = (const float*)d_in[37];

  float* ws = (float*)d_ws;
  size_t o = 0;
  float* wmean  = ws + o; o += (size_t)BT_ * LOD_;
  float* wvar   = ws + o; o += (size_t)BT_ * LOD_;
  float* ctrl   = ws + o; o += (size_t)BT_ * LSD_;
  float* prec   = ws + o; o += (size_t)B_ * LSD_;
  float* accm   = ws + o; o += (size_t)B_ * LSD_;
  float* taskm  = ws + o; o += (size_t)B_ * LSD_;
  float* noise  = ws + o; o += (size_t)B_ * LSD_;
  float* priorm = ws + o; o += (size_t)BT_ * LSD_;
  float* priorc = ws + o; o += (size_t)BT_ * 192;
  float* tmB    = ws + o; o += (size_t)4 * 16 * 4096;
  float* wc1p   = ws + o; o += (size_t)256 * 128;      // 2*LSD x HID packed
  float* wc2p   = ws + o; o += (size_t)128 * 16;       // HID x 16 packed (pad)

  k_init<<<(B_ * LSD_ + 255) / 256, 256, 0, stream>>>(prec, accm);
  k_prep<<<(4 * 16 * 4096 + 255) / 256, 256, 0, stream>>>(tm11, tm12, tm21, tm22, tmB);
  k_packw<<<(256 * 128 + 255) / 256, 256, 0, stream>>>(Wc1, wc1p, 256, 128, 128);
  k_packw<<<(128 * 16 + 255) / 256, 256, 0, stream>>>(Wc2, wc2p, 128, 15, 16);
  k_enc<<<BT_ / 16, 128, 0, stream>>>(obs, act, We1, be1, Wem, bem, Wev, bev,
                                      Wu1, bu1, Wu2, bu2, wmean, wvar, ctrl);
  k_ctx<<<(B_ * 63) / 16, 128, 0, stream>>>(obs, act, valid, Wt1, bt1,
                                            Wtm, btm, Wtv, btv, prec, accm);
  k_task<<<B_, 128, 0, stream>>>(prec, accm, Wtn, log_noise, taskm, noise);
  k_scan<<<B_ / 16, 512, 0, stream>>>(wmean, wvar, valid, ctrl, taskm, noise,
                                      wc1p, bc1, wc2p, bc2, tmB, priorm, priorc);
  k_dec<<<BT_ / 16, 128, 0, stream>>>(priorm, priorc, Wdm1, bdm1, Wdm2, bdm2,
                                      Wdv1, bdv1, Wdv2, bdv2, (float*)d_out);
}